// GAT_12936441496247
// MI455X (gfx1250) — compile-verified
//
#include <hip/hip_runtime.h>
#include <math.h>

#define NN      50000
#define EE      800000
#define IN_DIM_ 256
#define OUTD    64
#define NHEAD_  4
#define HD_     256
#define SLOPE_  0.2f
#define BN_EPS_ 1e-5f

#define KCHUNK  64
#define LDSPITCH 160   // words per K-pair row: 128 used + 32 pad (half-waves hit disjoint banks)

typedef __attribute__((ext_vector_type(2))) float v2f;
typedef __attribute__((ext_vector_type(8))) float v8f;

__device__ __forceinline__ float leaky_(float x) { return x >= 0.f ? x : SLOPE_ * x; }

// float atomic max via signed/unsigned bit trick (init with -inf)
__device__ __forceinline__ void atomic_max_f32(float* addr, float v) {
    if (v >= 0.f) atomicMax((int*)addr, __float_as_int(v));
    else          atomicMin((unsigned int*)addr, __float_as_uint(v));
}

// ---------------------------------------------------------------------------
// C[M,Ncol] = A[M,K] @ B[K,Ncol] + bias[Ncol]   via V_WMMA_F32_16X16X4_F32
// Block: 8 waves = 128 rows x 64 cols. B tile staged in LDS in K-pair-
// interleaved layout so each lane's B fragment is one ds_load_b64.
// Requires M % 16 == 0, K % KCHUNK == 0, Ncol % 64 == 0.
// ---------------------------------------------------------------------------
__global__ void __launch_bounds__(256)
wmma_gemm_bias(const float* __restrict__ A, const float* __restrict__ B,
               const float* __restrict__ bias, float* __restrict__ C,
               int M, int K, int Ncol)
{
    __shared__ float ldsB[(KCHUNK / 2) * LDSPITCH];   // 32*160*4 = 20.5 KB

    const int tid   = threadIdx.x;
    const int lane  = tid & 31;
    const int wave  = tid >> 5;
    const int mbase = blockIdx.y * 128 + wave * 16;
    const bool active = (mbase < M);
    const int nbase = blockIdx.x * 64;
    const int l15   = lane & 15;
    const int hw    = lane >> 4;          // 0 = lower half-wave, 1 = upper
    const int koff  = hw << 1;            // K offset 0 or 2 within the 16x4 A frag

    const int arow = active ? (mbase + l15) : l15;
    const float* Arow = A + (size_t)arow * K + koff;

    v8f acc0 = {}; v8f acc1 = {}; v8f acc2 = {}; v8f acc3 = {};

    for (int k0 = 0; k0 < K; k0 += KCHUNK) {
        __syncthreads();
        // cooperative fill: 32 K-pairs x 16 col-groups(4 cols) = 512 tasks
        for (int t = tid; t < 512; t += 256) {
            int kp = t >> 4;              // K-pair 0..31
            int cg = (t & 15) << 2;       // col 0,4,...,60
            const float* g0 = B + (size_t)(k0 + kp * 2)     * Ncol + nbase + cg;
            const float* g1 = B + (size_t)(k0 + kp * 2 + 1) * Ncol + nbase + cg;
            float4 r0 = *(const float4*)g0;
            float4 r1 = *(const float4*)g1;
            float* d = &ldsB[kp * LDSPITCH + cg * 2];
            v2f p0; p0.x = r0.x; p0.y = r1.x; ((v2f*)d)[0] = p0;
            v2f p1; p1.x = r0.y; p1.y = r1.y; ((v2f*)d)[1] = p1;
            v2f p2; p2.x = r0.z; p2.y = r1.z; ((v2f*)d)[2] = p2;
            v2f p3; p3.x = r0.w; p3.y = r1.w; ((v2f*)d)[3] = p3;
        }
        __syncthreads();

        #pragma unroll
        for (int kk = 0; kk < KCHUNK; kk += 4) {
            v2f a = *(const v2f*)(Arow + k0 + kk);               // A frag (global b64)
            const float* bb = &ldsB[((kk >> 1) + hw) * LDSPITCH + l15 * 2];
            v2f b0 = *(const v2f*)(bb);                          // ds_load_b64
            v2f b1 = *(const v2f*)(bb + 32);
            v2f b2 = *(const v2f*)(bb + 64);
            v2f b3 = *(const v2f*)(bb + 96);
            acc0 = __builtin_amdgcn_wmma_f32_16x16x4_f32(false, a, false, b0, (short)0, acc0, false, false);
            acc1 = __builtin_amdgcn_wmma_f32_16x16x4_f32(false, a, false, b1, (short)0, acc1, false, false);
            acc2 = __builtin_amdgcn_wmma_f32_16x16x4_f32(false, a, false, b2, (short)0, acc2, false, false);
            acc3 = __builtin_amdgcn_wmma_f32_16x16x4_f32(false, a, false, b3, (short)0, acc3, false, false);
        }
    }

    if (!active) return;
    const int rad = hw << 3;              // +8 rows for upper half-wave
    const float bi0 = bias[nbase + l15];
    const float bi1 = bias[nbase + 16 + l15];
    const float bi2 = bias[nbase + 32 + l15];
    const float bi3 = bias[nbase + 48 + l15];
    for (int r = 0; r < 8; ++r) {
        size_t row = (size_t)(mbase + rad + r) * Ncol;
        C[row + nbase      + l15] = acc0[r] + bi0;
        C[row + nbase + 16 + l15] = acc1[r] + bi1;
        C[row + nbase + 32 + l15] = acc2[r] + bi2;
        C[row + nbase + 48 + l15] = acc3[r] + bi3;
    }
}

// ---------------------------------------------------------------------------
// Init: rst = 0, m = -inf, denom = 0, colsum/colsumsq = 0
// ---------------------------------------------------------------------------
__global__ void init_ws_kernel(float* rst, float* m, float* denom,
                               float* colsum, float* colsumsq)
{
    int i = blockIdx.x * blockDim.x + threadIdx.x;
    if (i < NN * HD_) rst[i] = 0.f;
    if (i < NN * NHEAD_) { m[i] = -__builtin_inff(); denom[i] = 0.f; }
    if (i < OUTD) { colsum[i] = 0.f; colsumsq[i] = 0.f; }
}

// ---------------------------------------------------------------------------
// Per-edge GATv2 score: e[E,H] = attn . leaky(h_src[u] + h_dst[v]); atomic max into m.
// One wave per edge; lane owns 8 contiguous dims; head = lane/8.
// ---------------------------------------------------------------------------
__global__ void __launch_bounds__(256)
edge_scores(const float* __restrict__ hsrc, const float* __restrict__ hdst,
            const int* __restrict__ src, const int* __restrict__ dst,
            const float* __restrict__ attn, float* __restrict__ e,
            float* __restrict__ m)
{
    int edge = blockIdx.x * 8 + (threadIdx.x >> 5);
    if (edge >= EE) return;
    int lane = threadIdx.x & 31;
    int d0 = lane * 8;
    int u = src[edge], v = dst[edge];
    const float4* ps = (const float4*)(hsrc + (size_t)u * HD_ + d0);
    const float4* pd = (const float4*)(hdst + (size_t)v * HD_ + d0);
    const float4* pa = (const float4*)(attn + d0);      // attn flat [H*D]; d0 == h*64+dl
    float4 s0 = ps[0], s1 = ps[1];
    float4 t0 = pd[0], t1 = pd[1];
    float4 a0 = pa[0], a1 = pa[1];
    float s = a0.x * leaky_(s0.x + t0.x) + a0.y * leaky_(s0.y + t0.y)
            + a0.z * leaky_(s0.z + t0.z) + a0.w * leaky_(s0.w + t0.w)
            + a1.x * leaky_(s1.x + t1.x) + a1.y * leaky_(s1.y + t1.y)
            + a1.z * leaky_(s1.z + t1.z) + a1.w * leaky_(s1.w + t1.w);
    s += __shfl_xor(s, 1, 32);
    s += __shfl_xor(s, 2, 32);
    s += __shfl_xor(s, 4, 32);
    if ((lane & 7) == 0) {
        int h = lane >> 3;
        e[(size_t)edge * NHEAD_ + h] = s;
        atomic_max_f32(&m[(size_t)v * NHEAD_ + h], s);
    }
}

// ---------------------------------------------------------------------------
// ex = exp(e - m[dst]); accumulate denom[dst] (segment softmax pass 2)
// ---------------------------------------------------------------------------
__global__ void edge_exp(float* __restrict__ e, const int* __restrict__ dst,
                         const float* __restrict__ m, float* __restrict__ denom)
{
    int i = blockIdx.x * blockDim.x + threadIdx.x;
    if (i >= EE * NHEAD_) return;
    int edge = i >> 2, h = i & 3;
    int v = dst[edge];
    float ex = expf(e[i] - m[v * NHEAD_ + h]);
    e[i] = ex;
    atomicAdd(&denom[v * NHEAD_ + h], ex);
}

// ---------------------------------------------------------------------------
// rst[dst] += (ex/denom[dst]) * h_src[src]   (weighted scatter-add)
// ---------------------------------------------------------------------------
__global__ void __launch_bounds__(256)
edge_aggregate(const float* __restrict__ hsrc, const float* __restrict__ ex,
               const int* __restrict__ src, const int* __restrict__ dst,
               const float* __restrict__ denom, float* __restrict__ rst)
{
    int edge = blockIdx.x * 8 + (threadIdx.x >> 5);
    if (edge >= EE) return;
    int lane = threadIdx.x & 31;
    int h = lane >> 3;
    int d0 = lane * 8;
    int u = src[edge], v = dst[edge];
    float a = ex[(size_t)edge * NHEAD_ + h] / denom[(size_t)v * NHEAD_ + h];
    const float4* ps = (const float4*)(hsrc + (size_t)u * HD_ + d0);
    float4 s0 = ps[0], s1 = ps[1];
    float* out = rst + (size_t)v * HD_ + d0;
    atomicAdd(out + 0, a * s0.x);
    atomicAdd(out + 1, a * s0.y);
    atomicAdd(out + 2, a * s0.z);
    atomicAdd(out + 3, a * s0.w);
    atomicAdd(out + 4, a * s1.x);
    atomicAdd(out + 5, a * s1.y);
    atomicAdd(out + 6, a * s1.z);
    atomicAdd(out + 7, a * s1.w);
}

// ---------------------------------------------------------------------------
// bias2[c] = fc_b[c] + sum_k out_bias[k] * fc_W[k,c]   (folds out_bias thru fc)
// ---------------------------------------------------------------------------
__global__ void fold_bias(const float* __restrict__ out_bias,
                          const float* __restrict__ fc_W,
                          const float* __restrict__ fc_b,
                          float* __restrict__ bias2)
{
    int c = threadIdx.x;
    if (c >= OUTD) return;
    float s = fc_b[c];
    for (int k = 0; k < HD_; ++k) s += out_bias[k] * fc_W[k * OUTD + c];
    bias2[c] = s;
}

// ---------------------------------------------------------------------------
// BN stats: per-column sum and sum of squares over z[N,64]
// ---------------------------------------------------------------------------
__global__ void __launch_bounds__(256)
bn_stats(const float* __restrict__ z, float* __restrict__ colsum,
         float* __restrict__ colsumsq)
{
    int col = threadIdx.x & 63;
    int rg  = threadIdx.x >> 6;    // 0..3
    int base = blockIdx.x * 256;
    float s = 0.f, s2 = 0.f;
    for (int i = rg; i < 256; i += 4) {
        int row = base + i;
        if (row < NN) {
            float v = z[(size_t)row * OUTD + col];
            s += v; s2 += v * v;
        }
    }
    atomicAdd(&colsum[col], s);
    atomicAdd(&colsumsq[col], s2);
}

// ---------------------------------------------------------------------------
// out = leaky( (z - mu) * rsqrt(var+eps) * gamma + beta )
// ---------------------------------------------------------------------------
__global__ void bn_finalize(const float* __restrict__ z,
                            const float* __restrict__ colsum,
                            const float* __restrict__ colsumsq,
                            const float* __restrict__ gamma,
                            const float* __restrict__ beta,
                            float* __restrict__ out)
{
    int i = blockIdx.x * blockDim.x + threadIdx.x;
    if (i >= NN * OUTD) return;
    int col = i & 63;
    const float invN = 1.f / (float)NN;
    float mu  = colsum[col] * invN;
    float var = colsumsq[col] * invN - mu * mu;
    float y = (z[i] - mu) * rsqrtf(var + BN_EPS_) * gamma[col] + beta[col];
    out[i] = y >= 0.f ? y : SLOPE_ * y;
}

// ---------------------------------------------------------------------------
extern "C" void kernel_launch(void* const* d_in, const int* in_sizes, int n_in,
                              void* d_out, int out_size, void* d_ws, size_t ws_size,
                              hipStream_t stream) {
    const float* feat     = (const float*)d_in[0];
    const int*   src      = (const int*)  d_in[1];
    const int*   dst      = (const int*)  d_in[2];
    const float* W_src    = (const float*)d_in[3];
    const float* b_src    = (const float*)d_in[4];
    const float* W_dst    = (const float*)d_in[5];
    const float* b_dst    = (const float*)d_in[6];
    const float* attn     = (const float*)d_in[7];
    const float* out_bias = (const float*)d_in[8];
    const float* fc_W     = (const float*)d_in[9];
    const float* fc_b     = (const float*)d_in[10];
    const float* gamma    = (const float*)d_in[11];
    const float* beta     = (const float*)d_in[12];
    float* out = (float*)d_out;

    // workspace layout (float offsets)
    float* ws       = (float*)d_ws;
    float* h_src    = ws;                          // N*256 = 12,800,000
    float* h_dst    = h_src + (size_t)NN * HD_;    // 12,800,000
    float* rst      = h_dst + (size_t)NN * HD_;    // 12,800,000
    float* z        = rst   + (size_t)NN * HD_;    // N*64 = 3,200,000
    float* e        = z     + (size_t)NN * OUTD;   // E*4 = 3,200,000
    float* m        = e     + (size_t)EE * NHEAD_; // N*4 = 200,000
    float* denom    = m     + (size_t)NN * NHEAD_; // 200,000
    float* bias2    = denom + (size_t)NN * NHEAD_; // 64
    float* colsum   = bias2 + 64;                  // 64
    float* colsumsq = colsum + 64;                 // 64

    // 1) init accumulators
    init_ws_kernel<<<(NN * HD_ + 255) / 256, 256, 0, stream>>>(rst, m, denom, colsum, colsumsq);

    // 2) h_src = feat @ W_src + b_src ; h_dst = feat @ W_dst + b_dst  (WMMA f32)
    dim3 g1(HD_ / 64, (NN + 127) / 128);
    wmma_gemm_bias<<<g1, 256, 0, stream>>>(feat, W_src, b_src, h_src, NN, IN_DIM_, HD_);
    wmma_gemm_bias<<<g1, 256, 0, stream>>>(feat, W_dst, b_dst, h_dst, NN, IN_DIM_, HD_);

    // 3) edge scores + segment max
    edge_scores<<<(EE + 7) / 8, 256, 0, stream>>>(h_src, h_dst, src, dst, attn, e, m);

    // 4) exp + segment sum
    edge_exp<<<(EE * NHEAD_ + 255) / 256, 256, 0, stream>>>(e, dst, m, denom);

    // 5) weighted scatter-add aggregation
    edge_aggregate<<<(EE + 7) / 8, 256, 0, stream>>>(h_src, e, src, dst, denom, rst);

    // 6) fold out_bias into fc bias; z = rst @ fc_W + bias2  (WMMA f32)
    fold_bias<<<1, 64, 0, stream>>>(out_bias, fc_W, fc_b, bias2);
    dim3 g2(OUTD / 64, (NN + 127) / 128);
    wmma_gemm_bias<<<g2, 256, 0, stream>>>(rst, fc_W, bias2, z, NN, HD_, OUTD);

    // 7) batch-norm stats + finalize with LeakyReLU
    bn_stats<<<(NN + 255) / 256, 256, 0, stream>>>(z, colsum, colsumsq);
    bn_finalize<<<(NN * OUTD + 255) / 256, 256, 0, stream>>>(z, colsum, colsumsq, gamma, beta, out);
}